// NeuralKNN_56521769616034
// MI455X (gfx1250) — compile-verified
//
#include <hip/hip_runtime.h>
#include <hip/hip_bf16.h>
#include <math.h>

typedef __attribute__((ext_vector_type(2))) float v2f;
typedef __attribute__((ext_vector_type(8))) float v8f;

#define DE 64
#define NQ 256
#define NS 16384

// -------------------------------------------------------------------------
// Y[M x 64] = act( X[M x K] @ W[64 x K]^T + b ), ACT: 0 = exact gelu, 1 = sigmoid
// One wave computes 16 rows x 64 cols using 4 f32 WMMA accumulators.
// K is processed in 16-wide chunks: 20 batched loads, then 16 WMMAs, so loads
// for chunk c+1 overlap the WMMA burst of chunk c (no per-WMMA full waits).
// Block = 8 waves = 128 rows. M must be a multiple of 128 (16384 and 256 are).
// -------------------------------------------------------------------------
template <int KD, int ACT>
__global__ __launch_bounds__(256) void embed_layer_kernel(
    const float* __restrict__ X, const float* __restrict__ W,
    const float* __restrict__ bias, float* __restrict__ Y, int M)
{
    const int wave = threadIdx.x >> 5;
    const int lane = threadIdx.x & 31;
    const int half = lane >> 4;      // 0: lanes 0-15, 1: lanes 16-31
    const int l16  = lane & 15;
    const int row0 = (blockIdx.x * 8 + wave) * 16;
    if (row0 >= M) return;           // wave-uniform

    v8f acc0 = {0.f,0.f,0.f,0.f,0.f,0.f,0.f,0.f};
    v8f acc1 = acc0, acc2 = acc0, acc3 = acc0;

    const float* xrow = X + (size_t)(row0 + l16) * KD + half * 2;
    const float* w0   = W + (size_t)( 0 + l16) * KD + half * 2;
    const float* w1   = W + (size_t)(16 + l16) * KD + half * 2;
    const float* w2   = W + (size_t)(32 + l16) * KD + half * 2;
    const float* w3   = W + (size_t)(48 + l16) * KD + half * 2;

#pragma unroll
    for (int c = 0; c < KD / 16; ++c) {
        const int kc = c * 16;
        v2f a[4], f0[4], f1[4], f2[4], f3[4];
#pragma unroll
        for (int s = 0; s < 4; ++s) {          // batched fragment loads
            const int kb = kc + s * 4;
            a[s]  = *(const v2f*)(xrow + kb);
            f0[s] = *(const v2f*)(w0 + kb);
            f1[s] = *(const v2f*)(w1 + kb);
            f2[s] = *(const v2f*)(w2 + kb);
            f3[s] = *(const v2f*)(w3 + kb);
        }
#pragma unroll
        for (int s = 0; s < 4; ++s) {          // WMMA burst
            acc0 = __builtin_amdgcn_wmma_f32_16x16x4_f32(false, a[s], false, f0[s], (short)0, acc0, false, false);
            acc1 = __builtin_amdgcn_wmma_f32_16x16x4_f32(false, a[s], false, f1[s], (short)0, acc1, false, false);
            acc2 = __builtin_amdgcn_wmma_f32_16x16x4_f32(false, a[s], false, f2[s], (short)0, acc2, false, false);
            acc3 = __builtin_amdgcn_wmma_f32_16x16x4_f32(false, a[s], false, f3[s], (short)0, acc3, false, false);
        }
    }

    v8f accs[4] = {acc0, acc1, acc2, acc3};
#pragma unroll
    for (int nt = 0; nt < 4; ++nt) {
        const int col = nt * 16 + l16;
        const float bv = bias[col];
#pragma unroll
        for (int j = 0; j < 8; ++j) {
            const int row = row0 + j + half * 8;
            float v = accs[nt][j] + bv;
            float r;
            if (ACT == 0) {
                r = 0.5f * v * (1.0f + erff(v * 0.70710678118654752440f));
            } else {
                r = 1.0f / (1.0f + expf(-v));
            }
            Y[(size_t)row * DE + col] = r;
        }
    }
}

// -------------------------------------------------------------------------
// Row squared norms of an [M x 64] matrix.
// -------------------------------------------------------------------------
__global__ __launch_bounds__(256) void row_norm_kernel(
    const float* __restrict__ E, float* __restrict__ nrm, int M)
{
    const int r = blockIdx.x * blockDim.x + threadIdx.x;
    if (r >= M) return;
    const float4* p = (const float4*)(E + (size_t)r * DE);
    float s = 0.f;
#pragma unroll
    for (int i = 0; i < DE / 4; ++i) {
        float4 v = p[i];
        s += v.x * v.x + v.y * v.y + v.z * v.z + v.w * v.w;
    }
    nrm[r] = s;
}

// -------------------------------------------------------------------------
// Dist[q][s] = sqrt(max(||q||^2 + ||s||^2 - 2 q.s, 0)), with the reference's
// per-feature isclose -> +inf mask. All 32 K-fragments are preloaded (single
// wait), then 16 back-to-back WMMAs. The exact mask check only runs when
// dist <= 0.02 (safe necessary bound: exact closeness implies true dist
// <= 8*(atol+rtol) = 1.6e-4, and f32 cancellation noise in d2 is ~3e-5).
// grid = (NS/16/8, NQ/16), block = 256 (8 waves, each one 16x16 tile).
// -------------------------------------------------------------------------
__global__ __launch_bounds__(256) void dist_kernel(
    const float* __restrict__ Qe, const float* __restrict__ Se,
    const float* __restrict__ qn, const float* __restrict__ sn,
    float* __restrict__ Dist)
{
    const int wave = threadIdx.x >> 5;
    const int lane = threadIdx.x & 31;
    const int half = lane >> 4;
    const int l16  = lane & 15;
    const int qrow0 = blockIdx.y * 16;
    const int srow0 = (blockIdx.x * 8 + wave) * 16;

    const float* qrow = Qe + (size_t)(qrow0 + l16) * DE + half * 2;
    const float* srow = Se + (size_t)(srow0 + l16) * DE + half * 2;

    v2f a[16], b[16];
#pragma unroll
    for (int s = 0; s < 16; ++s) {             // all fragments in flight
        a[s] = *(const v2f*)(qrow + s * 4);
        b[s] = *(const v2f*)(srow + s * 4);
    }
    v8f acc = {0.f,0.f,0.f,0.f,0.f,0.f,0.f,0.f};
#pragma unroll
    for (int s = 0; s < 16; ++s) {             // WMMA burst
        acc = __builtin_amdgcn_wmma_f32_16x16x4_f32(false, a[s], false, b[s], (short)0, acc, false, false);
    }

    const int col = srow0 + l16;
    const float ssn = sn[col];
#pragma unroll
    for (int j = 0; j < 8; ++j) {
        const int row = qrow0 + j + half * 8;
        float d2 = qn[row] + ssn - 2.0f * acc[j];
        float dist = sqrtf(fmaxf(d2, 0.0f));
        if (dist <= 0.02f) {   // rare: exact reference isclose check
            const float* qp = Qe + (size_t)row * DE;
            const float* sp = Se + (size_t)col * DE;
            bool close = true;
            for (int d = 0; d < DE; ++d) {
                float sv = sp[d];
                if (fabsf(qp[d] - sv) > (1e-5f + 1e-5f * fabsf(sv))) { close = false; break; }
            }
            if (close) dist = __builtin_inff();
        }
        Dist[(size_t)row * NS + col] = dist;
    }
}

// -------------------------------------------------------------------------
// Per-query top-32 smallest distances + softmax(-d/0.1) weighted label sum.
// One block per query. Each thread owns 64 distances (strided, coalesced) in
// registers with a 64-bit removal mask; 32 rounds of block argmin via LDS.
// -------------------------------------------------------------------------
__global__ __launch_bounds__(256) void topk_kernel(
    const float* __restrict__ Dist, const float* __restrict__ labels,
    float* __restrict__ out)
{
    __shared__ float sv[256];
    __shared__ int   si[256];
    __shared__ float kd[32];
    __shared__ int   ki[32];

    const int q = blockIdx.x;
    const int t = threadIdx.x;
    const float* drow = Dist + (size_t)q * NS;

    float v[64];
#pragma unroll
    for (int i = 0; i < 64; ++i) v[i] = drow[t + i * 256];

    unsigned long long removed = 0ull;
    for (int k = 0; k < 32; ++k) {
        float lmin = __builtin_inff();
        int lidx = 0;
#pragma unroll
        for (int i = 0; i < 64; ++i) {
            bool rem = (removed >> i) & 1ull;
            if (!rem && v[i] < lmin) { lmin = v[i]; lidx = i; }
        }
        sv[t] = lmin;
        si[t] = t + lidx * 256;
        __syncthreads();
        for (int s = 128; s > 0; s >>= 1) {
            if (t < s) {
                if (sv[t + s] < sv[t]) { sv[t] = sv[t + s]; si[t] = si[t + s]; }
            }
            __syncthreads();
        }
        const int   widx = si[0];
        const float wval = sv[0];
        if (t == 0) { kd[k] = wval; ki[k] = widx; }
        if ((widx & 255) == t) removed |= 1ull << (widx >> 8);
        __syncthreads();
    }

    if (t == 0) {
        const float dmin = kd[0];   // repeated-min -> kd[0] is the smallest
        float wsum = 0.f, accv = 0.f;
        for (int k = 0; k < 32; ++k) {
            float w = expf((dmin - kd[k]) * 10.0f);  // 1/T = 10
            wsum += w;
            accv += w * labels[ki[k]];
        }
        out[q] = accv / wsum;
    }
}

// -------------------------------------------------------------------------
extern "C" void kernel_launch(void* const* d_in, const int* in_sizes, int n_in,
                              void* d_out, int out_size, void* d_ws, size_t ws_size,
                              hipStream_t stream) {
    (void)in_sizes; (void)n_in; (void)out_size; (void)ws_size;
    const float* x      = (const float*)d_in[0];   // [256,256]
    const float* sx     = (const float*)d_in[1];   // [16384,256]
    const float* slab   = (const float*)d_in[2];   // [16384,1]
    const float* W1     = (const float*)d_in[3];   // [64,256]
    const float* b1     = (const float*)d_in[4];
    const float* W2     = (const float*)d_in[5];   // [64,64]
    const float* b2     = (const float*)d_in[6];
    const float* W3     = (const float*)d_in[7];   // [64,64]
    const float* b3     = (const float*)d_in[8];
    float* out = (float*)d_out;

    float* ws   = (float*)d_ws;
    float* bufA = ws;                          // 16384*64
    float* bufB = bufA + (size_t)NS * DE;      // 16384*64
    float* qA   = bufB + (size_t)NS * DE;      // 256*64
    float* qB   = qA   + (size_t)NQ * DE;      // 256*64
    float* sn   = qB   + (size_t)NQ * DE;      // 16384
    float* qn   = sn   + NS;                   // 256
    float* Dst  = qn   + NQ;                   // 256*16384

    // Support embedding: L1 (gelu) -> L2 (gelu) -> L3 (sigmoid)
    embed_layer_kernel<256, 0><<<NS / 128, 256, 0, stream>>>(sx,   W1, b1, bufA, NS);
    embed_layer_kernel<64,  0><<<NS / 128, 256, 0, stream>>>(bufA, W2, b2, bufB, NS);
    embed_layer_kernel<64,  1><<<NS / 128, 256, 0, stream>>>(bufB, W3, b3, bufA, NS);
    // Query embedding
    embed_layer_kernel<256, 0><<<NQ / 128, 256, 0, stream>>>(x,  W1, b1, qA, NQ);
    embed_layer_kernel<64,  0><<<NQ / 128, 256, 0, stream>>>(qA, W2, b2, qB, NQ);
    embed_layer_kernel<64,  1><<<NQ / 128, 256, 0, stream>>>(qB, W3, b3, qA, NQ);
    // Row norms
    row_norm_kernel<<<NS / 256, 256, 0, stream>>>(bufA, sn, NS);
    row_norm_kernel<<<1,        256, 0, stream>>>(qA,   qn, NQ);
    // Distance matrix (WMMA) with isclose mask
    dim3 dg(NS / 16 / 8, NQ / 16);
    dist_kernel<<<dg, 256, 0, stream>>>(qA, bufA, qn, sn, Dst);
    // Top-k + softmax-weighted label sum
    topk_kernel<<<NQ, 256, 0, stream>>>(Dst, slab, out);
}